// SparseDetailLevel_61993557950727
// MI455X (gfx1250) — compile-verified
//
#include <hip/hip_runtime.h>
#include <hip/hip_bf16.h>
#include <stdint.h>

// SparseDetailLevel gather: out[j,0,c,x,y,z] = occupied ? tiles[rank,j,c,ox,oy,oz] : 0
// Window [8,72) over 8^3 tiles of 16^3 -> 5^3 candidate tiles, rank = prefix count.
//
// Memory-bound copy: ~205MB write + <=205MB read -> ~18us floor at 23.3 TB/s.
// CDNA5 paths used: global_load_async_to_lds_b128 (+ s_wait_asynccnt) for the
// per-block table stage, non-temporal b128 loads/stores for the streamed data.

typedef float v4f __attribute__((ext_vector_type(4)));

#define BLOCKV 16
#define TGRID  8
#define NTW    5          // tiles per axis overlapping the window
#define WOFF   8          // s0 - t_start*16
#define NJC    196u       // 7 * 28
#define TILE_STRIDE  802816u   // 7*28*4096
#define J_STRIDE     114688u   // 28*4096
#define C_STRIDE     4096u
#define GROUPS_TOTAL 12845056u // 196*64*64*16 float4 groups
#define MAIN_BLOCKS  12544
#define ITERS        4
#define ITER_STRIDE  3211264u  // MAIN_BLOCKS*256 == 49*65536 (spatial bits invariant)

// ---------------------------------------------------------------------------
// Kernel 1: build 5x5x5 table of occupied-tile ranks (or -1) into d_ws.
// 128 threads, one block. Entries 125..127 padded (-1) so the 512B async
// load in kernel 2 reads defined data.
// ---------------------------------------------------------------------------
__global__ void sdl_build_table(const unsigned char* __restrict__ occ,
                                int* __restrict__ tbl) {
    __shared__ unsigned char s_occ[512];
    const int t = threadIdx.x;                       // 0..127
    ((uint32_t*)s_occ)[t] = ((const uint32_t*)occ)[t]; // 128*4 = 512 bytes
    __syncthreads();
    if (t >= 125) { tbl[t] = -1; return; }
    const int gx = t / 25;
    const int r  = t - gx * 25;
    const int gy = r / 5;
    const int gz = r - gy * 5;
    const int lin = (gx * TGRID + gy) * TGRID + gz;  // window starts at tile 0
    int cnt = 0;
    for (int i = 0; i < lin; ++i) cnt += s_occ[i] ? 1 : 0;
    tbl[t] = s_occ[lin] ? cnt : -1;
}

// ---------------------------------------------------------------------------
// Kernel 2: streamed gather. One float4 group per thread per iteration.
//   flat group id g: [jc:8][x:6][y:6][z4:4]  (jc = j*28+c)
// Grid stride is 49*65536, so x/y/z4 (and thus the whole spatial part of the
// source offset + table entry) are loop-invariant; only jc changes.
// ---------------------------------------------------------------------------
__global__ void __launch_bounds__(256)
sdl_gather(const float* __restrict__ tiles,
           float* __restrict__ out,
           const int* __restrict__ tbl_g) {
    __shared__ int s_tbl[128];

    // Stage the 512B rank table into LDS with the CDNA5 async-copy path.
    // First wave (threads 0..31, EXEC all ones) issues 32 lanes x 16B = 512B.
    if (threadIdx.x < 32) {
        uint32_t lds_addr = (uint32_t)(uintptr_t)(&s_tbl[0]) + threadIdx.x * 16u;
        uint64_t gaddr    = (uint64_t)(uintptr_t)tbl_g + threadIdx.x * 16u;
        asm volatile(
            "global_load_async_to_lds_b128 %0, %1, off\n\t"
            "s_wait_asynccnt 0x0"
            :
            : "v"(lds_addr), "v"(gaddr), "r"(s_tbl)
            : "memory");
    }
    __syncthreads();

    const uint32_t g0 = blockIdx.x * 256u + threadIdx.x;   // < ITER_STRIDE
    const uint32_t z4  = g0 & 15u;
    const uint32_t y   = (g0 >> 4) & 63u;
    const uint32_t x   = (g0 >> 10) & 63u;
    const uint32_t jc0 = g0 >> 16;                          // 0..48

    const uint32_t gx = x + WOFF, gy = y + WOFF, gz = WOFF + 4u * z4;
    const uint32_t tx = gx >> 4, ox = gx & 15u;
    const uint32_t ty = gy >> 4, oy = gy & 15u;
    const uint32_t tz = gz >> 4, oz = gz & 15u;

    const int tile = s_tbl[(tx * NTW + ty) * NTW + tz];
    // spatial part of the source offset (elements); float4 never crosses a
    // tile row boundary because oz is 4-aligned and rows are 16 wide.
    const uint32_t soff = ox * 256u + oy * 16u + oz;
    const float* src_base =
        (tile >= 0) ? tiles + (size_t)(uint32_t)tile * TILE_STRIDE + soff : nullptr;

    v4f* __restrict__ out4 = (v4f*)out;

#pragma unroll
    for (int i = 0; i < ITERS; ++i) {
        const uint32_t jc = jc0 + 49u * i;                  // < 196
        const uint32_t j  = jc / 28u;
        const uint32_t c  = jc - 28u * j;
        v4f v = {0.f, 0.f, 0.f, 0.f};
        if (tile >= 0) {
            const v4f* src = (const v4f*)(src_base + j * J_STRIDE + c * C_STRIDE);
            v = __builtin_nontemporal_load(src);            // th:NT (read-once)
        }
        __builtin_nontemporal_store(v, out4 + g0 + (uint32_t)i * ITER_STRIDE);
    }
}

// ---------------------------------------------------------------------------
extern "C" void kernel_launch(void* const* d_in, const int* in_sizes, int n_in,
                              void* d_out, int out_size, void* d_ws, size_t ws_size,
                              hipStream_t stream) {
    const float* tiles        = (const float*)d_in[0];         // (154,7,28,16,16,16) f32
    const unsigned char* occ  = (const unsigned char*)d_in[1]; // (8,8,8) bool (1B each)
    float* out                = (float*)d_out;                 // (7,1,28,64,64,64) f32
    int* tbl                  = (int*)d_ws;                    // 128 ints scratch

    sdl_build_table<<<1, 128, 0, stream>>>(occ, tbl);
    sdl_gather<<<MAIN_BLOCKS, 256, 0, stream>>>(tiles, out, tbl);
}